// PolicyHead_8014408974365
// MI455X (gfx1250) — compile-verified
//
#include <hip/hip_runtime.h>

typedef __attribute__((ext_vector_type(2))) float v2f;
typedef __attribute__((ext_vector_type(8))) float v8f;

#define N_NODES 50000
#define N_EDGES 1600000
#define D_IN 128
#define D_E 64
#define NODE_TILES (N_NODES / 16)   // 3125
#define EDGE_TILES (N_EDGES / 16)   // 100000

// D = A(16x4 f32) * B(4x16 f32) + C(16x16 f32), exact f32 matrix pipe.
__device__ __forceinline__ v8f wmma_f32(v2f a, v2f b, v8f c) {
  return __builtin_amdgcn_wmma_f32_16x16x4_f32(
      /*neg_a=*/false, a, /*neg_b=*/false, b,
      /*c_mod=*/(short)0, c, /*reuse_a=*/false, /*reuse_b=*/false);
}

// ---------------------------------------------------------------------------
// Kernel 1: P[m][0:64]   = emb[m] @ W1[0:128]        (src projection)
//           P[m][64:128] = emb[m] @ W1[128:256]      (dst projection)
// Each wave owns a 32-column group (2 n-tiles), preloads B frags in regs,
// grid-strides over 16-node M tiles.
// ---------------------------------------------------------------------------
__global__ __launch_bounds__(256) void node_proj_kernel(
    const float* __restrict__ emb, const float* __restrict__ W1,
    float* __restrict__ P, int nWaveGroups) {
  const int lane = threadIdx.x & 31;
  const int half = lane >> 4;     // 0: K pair base 0, 1: K pair base 2
  const int l16  = lane & 15;
  const int wave = blockIdx.x * (blockDim.x >> 5) + (threadIdx.x >> 5);
  const int cg   = wave & 3;      // 32-column group
  const int wg   = wave >> 2;

  // B frags: B[k][c] = W1[(c<64 ? k : 128+k)*64 + (c%64)]
  // lane -> col c = cg*32 + tt*16 + l16 ; element i -> k = ks*4 + half*2 + i
  v2f Bf[2][32];
#pragma unroll
  for (int tt = 0; tt < 2; ++tt) {
    const int c = cg * 32 + tt * 16 + l16;
    const float* wcol = W1 + ((c < 64) ? c : (128 * 64 + (c - 64)));
#pragma unroll
    for (int ks = 0; ks < 32; ++ks) {
      const int k0 = ks * 4 + half * 2;
      v2f b;
      b.x = wcol[k0 * 64];
      b.y = wcol[(k0 + 1) * 64];
      Bf[tt][ks] = b;
    }
  }

  for (int mt = wg; mt < NODE_TILES; mt += nWaveGroups) {
    // A frag: lane row = l16, element i -> k = ks*4 + half*2 + i (8B aligned)
    const float* arow = emb + (size_t)(mt * 16 + l16) * D_IN + half * 2;
    v8f acc0 = {};
    v8f acc1 = {};
#pragma unroll
    for (int ks = 0; ks < 32; ++ks) {
      v2f a = *(const v2f*)(arow + ks * 4);
      acc0 = wmma_f32(a, Bf[0][ks], acc0);
      acc1 = wmma_f32(a, Bf[1][ks], acc1);
    }
    // C layout: element j, lane -> row m = j + 8*half, col = l16
#pragma unroll
    for (int j = 0; j < 8; ++j) {
      const int m = mt * 16 + j + 8 * half;
      float* prow = P + (size_t)m * 128 + cg * 32 + l16;
      prow[0]  = acc0[j];
      prow[16] = acc1[j];
    }
  }
}

// ---------------------------------------------------------------------------
// Kernel 2: per 16-edge tile:
//   h(16x64) = edge_feats_tile @ W1_e   (WMMA, 4 n-tiles x 16 k-steps)
//   h += P[src][n] + P[dst][64+n] + b1 ; relu ; logits = h @ W2 + b2
// Gathers use uniform-base + unsigned 32-bit byte offsets (GVS addressing).
// ---------------------------------------------------------------------------
__global__ __launch_bounds__(256) void edge_kernel(
    const float* __restrict__ P, const int* __restrict__ src_idx,
    const int* __restrict__ dst_idx, const float* __restrict__ ef,
    const float* __restrict__ W1, const float* __restrict__ b1,
    const float* __restrict__ W2, const float* __restrict__ b2,
    float* __restrict__ out, int nWaves) {
  const int lane = threadIdx.x & 31;
  const int half = lane >> 4;
  const int l16  = lane & 15;
  const int wave = blockIdx.x * (blockDim.x >> 5) + (threadIdx.x >> 5);

  // W1_e B frags (loop-invariant, kept in VGPRs): B[k][c] = W1[(256+k)*64 + c]
  v2f Bf[4][16];
#pragma unroll
  for (int tt = 0; tt < 4; ++tt) {
    const float* wcol = W1 + 256 * 64 + tt * 16 + l16;
#pragma unroll
    for (int ks = 0; ks < 16; ++ks) {
      const int k0 = ks * 4 + half * 2;
      v2f b;
      b.x = wcol[k0 * 64];
      b.y = wcol[(k0 + 1) * 64];
      Bf[tt][ks] = b;
    }
  }
  float b1v[4], w2v[4];
#pragma unroll
  for (int t = 0; t < 4; ++t) {
    b1v[t] = b1[t * 16 + l16];
    w2v[t] = W2[t * 16 + l16];
  }
  const float b2s = b2[0];

  const char* Pb = (const char*)P;
  const unsigned ncb0 = (unsigned)l16 * 4u;  // per-lane column byte offset

  for (int et = wave; et < EDGE_TILES; et += nWaves) {
    const int e0 = et * 16;
    const float* arow = ef + (size_t)(e0 + l16) * D_E + half * 2;

    // Prefetch next tile's contiguous 4KB edge_feats block (32 lanes x 128B).
    if (et + nWaves < EDGE_TILES) {
      const char* nt = (const char*)ef +
                       (size_t)(e0 + nWaves * 16) * (D_E * 4) + lane * 128;
      __builtin_prefetch(nt, 0, 0);
    }

    v8f acc0 = {};
    v8f acc1 = {};
    v8f acc2 = {};
    v8f acc3 = {};
#pragma unroll
    for (int ks = 0; ks < 16; ++ks) {
      v2f a = *(const v2f*)(arow + ks * 4);
      acc0 = wmma_f32(a, Bf[0][ks], acc0);
      acc1 = wmma_f32(a, Bf[1][ks], acc1);
      acc2 = wmma_f32(a, Bf[2][ks], acc2);
      acc3 = wmma_f32(a, Bf[3][ks], acc3);
    }

    // Rows owned by this lane: m = j + 8*half (same set for all 16 lanes of a half)
    // 32-bit unsigned byte offsets into P (P is 25.6MB, fits easily).
    unsigned soff[8], doff[8];
#pragma unroll
    for (int j = 0; j < 8; ++j) {
      const int m = e0 + j + 8 * half;
      soff[j] = (unsigned)src_idx[m] * 512u + ncb0;          // row*128 floats
      doff[j] = (unsigned)dst_idx[m] * 512u + 256u + ncb0;   // + dst half
    }

    float part[8];
#pragma unroll
    for (int j = 0; j < 8; ++j) part[j] = 0.f;

#pragma unroll
    for (int t = 0; t < 4; ++t) {
      const unsigned tb = (unsigned)t * 64u;  // 16 cols * 4B
      const v8f accv = (t == 0) ? acc0 : (t == 1) ? acc1 : (t == 2) ? acc2 : acc3;
#pragma unroll
      for (int j = 0; j < 8; ++j) {
        const float gs = *(const float*)(Pb + (soff[j] + tb));
        const float gd = *(const float*)(Pb + (doff[j] + tb));
        float h = accv[j] + (gs + gd) + b1v[t];
        h = fmaxf(h, 0.f);
        part[j] = fmaf(h, w2v[t], part[j]);
      }
    }

    // Reduce over the 16 lanes of each half (xor masks stay within bits 0..3)
#pragma unroll
    for (int j = 0; j < 8; ++j) {
#pragma unroll
      for (int off = 1; off < 16; off <<= 1)
        part[j] += __shfl_xor(part[j], off, 32);
    }

    if (l16 == 0) {  // lane 0 -> rows 0..7, lane 16 -> rows 8..15
      float* o = out + e0 + half * 8;
#pragma unroll
      for (int j = 0; j < 8; ++j) o[j] = part[j] + b2s;
    }
  }
}

extern "C" void kernel_launch(void* const* d_in, const int* in_sizes, int n_in,
                              void* d_out, int out_size, void* d_ws, size_t ws_size,
                              hipStream_t stream) {
  (void)in_sizes; (void)n_in; (void)out_size; (void)ws_size;
  const float* emb     = (const float*)d_in[0];
  const int*   src_idx = (const int*)d_in[1];
  const int*   dst_idx = (const int*)d_in[2];
  const float* ef      = (const float*)d_in[3];
  const float* W1      = (const float*)d_in[4];
  const float* b1      = (const float*)d_in[5];
  const float* W2      = (const float*)d_in[6];
  const float* b2      = (const float*)d_in[7];
  float* out = (float*)d_out;
  float* P   = (float*)d_ws;  // 50000 * 128 floats = 25.6 MB scratch

  const int K1B = 256;   // 2048 waves -> 512 wave-groups per column group
  node_proj_kernel<<<K1B, 256, 0, stream>>>(emb, W1, P, (K1B * 8) / 4);

  const int K2B = 1024;  // 8192 waves, ~12 edge tiles each
  edge_kernel<<<K2B, 256, 0, stream>>>(P, src_idx, dst_idx, ef, W1, b1, W2, b2,
                                       out, K2B * 8);
}